// SiameseSpectralModel_42545946034790
// MI455X (gfx1250) — compile-verified
//
#include <hip/hip_runtime.h>
#include <math.h>

typedef __attribute__((ext_vector_type(2))) float v2f;
typedef __attribute__((ext_vector_type(8))) float v8f;

// Problem constants (from reference)
#define NSPEC   512      // B*2 spectra
#define NPEAK   512      // P
#define NGROUP  3333     // GROUPS = 100000 // 30
#define XVALID  9999     // BINNER_OUT
#define XCOLS   10000    // padded K for GEMM0 (multiple of 16, last col zero)
#define H0COLS  1008     // 1000 padded to multiple of 16
#define EMBD    400

// ---------------------------------------------------------------------------
// Stage 0: initialize X[b, g*3+o] = binner_b[g,o]; padded col = 0
// ---------------------------------------------------------------------------
__global__ __launch_bounds__(256) void init_x_kernel(
    const float* __restrict__ binner_b, float* __restrict__ X) {
  int idx = blockIdx.x * 256 + threadIdx.x;   // 512*10000 total
  int col = idx % XCOLS;
  X[idx] = (col < XVALID) ? binner_b[col] : 0.0f;
}

// ---------------------------------------------------------------------------
// Stage 1: fused binning + grouped binner einsum via atomics.
// Each peak lands in bin idx -> group g, in-group i; contributes
// inten^0.5 * binner_w[g,i,o] to X[spec, g*3+o], o=0..2.
// ---------------------------------------------------------------------------
__global__ __launch_bounds__(256) void bin_scatter_kernel(
    const float* __restrict__ mz, const float* __restrict__ inten,
    const float* __restrict__ binner_w, float* __restrict__ X) {
  int s = blockIdx.x;          // spectrum 0..511
  for (int p = threadIdx.x; p < NPEAK; p += 256) {
    float m = mz[s * NPEAK + p];
    if (m >= 0.0f && m < 1000.0f) {
      int idx = (int)(m / 0.01f);
      idx = min(max(idx, 0), 99999);
      int g = idx / 30;
      if (g < NGROUP) {              // bins >= 99990 are sliced off in ref
        int i = idx - g * 30;
        float v = sqrtf(inten[s * NPEAK + p]);   // inten ** 0.5
        const float* w = binner_w + (g * 30 + i) * 3;
        float* xr = X + s * XCOLS + g * 3;
        atomicAdd(xr + 0, v * w[0]);
        atomicAdd(xr + 1, v * w[1]);
        atomicAdd(xr + 2, v * w[2]);
      }
    }
  }
}

// ---------------------------------------------------------------------------
// Stage 2: tiled f32 WMMA GEMM  C = act(A @ B + bias)
//   - workgroup 256 thr = 8 waves; block tile 64(M) x 128(N); KB = 16
//   - waves 2(M) x 4(N); each wave register-tiles 32x32 (4 accumulators);
//     per K-step: 2 A-frags + 2 B-frags (four ds_load_b64) feed 4 WMMAs
//   - LDS staging is DOUBLE-BUFFERED via async DMA:
//       global_load_async_to_lds_b32 (ASYNCcnt) stages block kb+1 while
//       WMMAs consume block kb; s_wait_asynccnt 0 + barrier rotates buffers
//   - Bs is stored transposed (Bs[n][k], even stride) so every fragment
//     (k,k+1) is one aligned 64-bit LDS load -> no register repacking
//   - requires lda == K (callers pad A with zero columns); B loads guarded
//     by k < KvalidB && n < N, zero-filled otherwise
// ---------------------------------------------------------------------------
__global__ __launch_bounds__(256) void wmma_gemm_kernel(
    const float* __restrict__ A, const float* __restrict__ B,
    const float* __restrict__ bias, float* __restrict__ C,
    int M, int N, int Npad, int K, int KvalidB,
    int lda, int ldb, int ldc, int do_relu) {
  __shared__ float As[2][64][18];    // [buf][row][k]  (stride 18: 8B-aligned pairs)
  __shared__ float Bs[2][128][18];   // [buf][col][k]  transposed

  int t    = threadIdx.x;
  int l    = t & 31;            // lane in wave32
  int w    = t >> 5;            // wave 0..7
  int wm   = w >> 2;            // 0..1  (32-row sub-tile)
  int wn   = w & 3;             // 0..3  (32-col sub-tile)
  int l15  = l & 15;
  int klo  = (l >> 4) << 1;     // 0 for lanes 0-15, 2 for lanes 16-31

  int tileM = blockIdx.y << 6;  // * 64
  int tileN = blockIdx.x << 7;  // * 128

  v8f acc00 = {0.f,0.f,0.f,0.f,0.f,0.f,0.f,0.f};
  v8f acc01 = acc00, acc10 = acc00, acc11 = acc00;

  // async-DMA stage of one 64x16 A tile + 16x128 B tile into buffer `buf`
  auto stage = [&](int kb0, int buf) {
    // A tile: 64 rows x 16 k, always in-bounds (lda == K, M tiles exact)
    for (int e = t; e < 64 * 16; e += 256) {
      int r = e >> 4, k = e & 15;
      unsigned dst = (unsigned)(unsigned long long)&As[buf][r][k];
      const float* src = A + (tileM + r) * lda + (kb0 + k);
      asm volatile("global_load_async_to_lds_b32 %0, %1, off"
                   :: "v"(dst), "v"(src) : "memory");
    }
    // B tile: 16 k x 128 n, written transposed; guarded
    for (int e = t; e < 16 * 128; e += 256) {
      int kr = e >> 7, n = e & 127;
      int kg = kb0 + kr, ng = tileN + n;
      if (kg < KvalidB && ng < N) {
        unsigned dst = (unsigned)(unsigned long long)&Bs[buf][n][kr];
        const float* src = B + kg * ldb + ng;
        asm volatile("global_load_async_to_lds_b32 %0, %1, off"
                     :: "v"(dst), "v"(src) : "memory");
      } else {
        Bs[buf][n][kr] = 0.0f;
      }
    }
  };

  int nkb = K >> 4;             // all callers use K % 16 == 0
  stage(0, 0);
  asm volatile("s_wait_asynccnt 0x0" ::: "memory");
  __syncthreads();

  for (int kb = 0; kb < nkb; ++kb) {
    int cur = kb & 1;
    if (kb + 1 < nkb) stage((kb + 1) << 4, cur ^ 1);   // overlap DMA w/ WMMA

#pragma unroll
    for (int kk = 0; kk < 16; kk += 4) {
      v2f a0 = *(const v2f*)&As[cur][wm * 32 +      l15][kk + klo];
      v2f a1 = *(const v2f*)&As[cur][wm * 32 + 16 + l15][kk + klo];
      v2f b0 = *(const v2f*)&Bs[cur][wn * 32 +      l15][kk + klo];
      v2f b1 = *(const v2f*)&Bs[cur][wn * 32 + 16 + l15][kk + klo];
      acc00 = __builtin_amdgcn_wmma_f32_16x16x4_f32(
          false, a0, false, b0, (short)0, acc00, false, false);
      acc01 = __builtin_amdgcn_wmma_f32_16x16x4_f32(
          false, a0, false, b1, (short)0, acc01, false, false);
      acc10 = __builtin_amdgcn_wmma_f32_16x16x4_f32(
          false, a1, false, b0, (short)0, acc10, false, false);
      acc11 = __builtin_amdgcn_wmma_f32_16x16x4_f32(
          false, a1, false, b1, (short)0, acc11, false, false);
    }

    asm volatile("s_wait_asynccnt 0x0" ::: "memory");  // next buffer staged
    __syncthreads();                                   // all readers done
  }

  // store: lane holds N=l15 of its 16-col group, rows r + (l>=16 ? 8 : 0)
  int rsel = (l >> 4) << 3;
#pragma unroll
  for (int i = 0; i < 2; ++i) {
    int rbase = tileM + wm * 32 + i * 16 + rsel;
#pragma unroll
    for (int j = 0; j < 2; ++j) {
      int col = tileN + wn * 32 + j * 16 + l15;
      if (col < Npad) {
        const v8f& acc = i == 0 ? (j == 0 ? acc00 : acc01)
                                : (j == 0 ? acc10 : acc11);
#pragma unroll
        for (int r = 0; r < 8; ++r) {
          float v = 0.0f;
          if (col < N) {
            v = acc[r] + bias[col];
            if (do_relu) v = fmaxf(v, 0.0f);
          }
          C[(rbase + r) * ldc + col] = v;
        }
      }
    }
  }
}

// ---------------------------------------------------------------------------
// Stage 3: cosine similarity per pair; one wave32 per pair.
// ---------------------------------------------------------------------------
__global__ __launch_bounds__(256) void cosine_kernel(
    const float* __restrict__ E, float* __restrict__ out) {
  int l = threadIdx.x & 31;
  int w = threadIdx.x >> 5;
  int pair = blockIdx.x * 8 + w;              // 256 pairs total
  const float* e1 = E + (2 * pair) * EMBD;
  const float* e2 = e1 + EMBD;
  float d = 0.f, s1 = 0.f, s2 = 0.f;
  for (int j = l; j < EMBD; j += 32) {
    float a = e1[j], b = e2[j];
    d += a * b; s1 += a * a; s2 += b * b;
  }
#pragma unroll
  for (int off = 16; off > 0; off >>= 1) {
    d  += __shfl_xor(d,  off, 32);
    s1 += __shfl_xor(s1, off, 32);
    s2 += __shfl_xor(s2, off, 32);
  }
  if (l == 0)
    out[pair] = d / (fmaxf(sqrtf(s1), 1e-6f) * fmaxf(sqrtf(s2), 1e-6f));
}

// ---------------------------------------------------------------------------
extern "C" void kernel_launch(void* const* d_in, const int* in_sizes, int n_in,
                              void* d_out, int out_size, void* d_ws, size_t ws_size,
                              hipStream_t stream) {
  const float* mz       = (const float*)d_in[0];
  const float* inten    = (const float*)d_in[1];
  const float* binner_w = (const float*)d_in[2];
  const float* binner_b = (const float*)d_in[3];
  const float* w0 = (const float*)d_in[4];
  const float* b0 = (const float*)d_in[5];
  const float* w1 = (const float*)d_in[6];
  const float* b1 = (const float*)d_in[7];
  const float* w2 = (const float*)d_in[8];
  const float* b2 = (const float*)d_in[9];
  const float* we = (const float*)d_in[10];
  const float* be = (const float*)d_in[11];

  float* ws = (float*)d_ws;
  float* X  = ws;                          // 512 x 10000 (padded)
  float* H0 = X  + NSPEC * XCOLS;          // 512 x 1008  (padded)
  float* H1 = H0 + NSPEC * H0COLS;         // 512 x 800
  float* H2 = H1 + NSPEC * 800;            // 512 x 800
  float* E  = H2 + NSPEC * 800;            // 512 x 400

  // Stage 0+1: binner output via fused scatter
  init_x_kernel<<<(NSPEC * XCOLS) / 256, 256, 0, stream>>>(binner_b, X);
  bin_scatter_kernel<<<NSPEC, 256, 0, stream>>>(mz, inten, binner_w, X);

  // Stage 2: MLP (block tile 64x128, grid = (ceil(Npad/128), M/64))
  // H0 = relu(X @ w0 + b0):  M=512 N=1000 K=9999 (A padded to 10000)
  wmma_gemm_kernel<<<dim3((1008 + 127) / 128, 8), 256, 0, stream>>>(
      X, w0, b0, H0, 512, 1000, 1008, 10000, 9999, XCOLS, 1000, H0COLS, 1);
  // H1 = relu(H0 @ w1 + b1): M=512 N=800 K=1000 (A padded to 1008)
  wmma_gemm_kernel<<<dim3((800 + 127) / 128, 8), 256, 0, stream>>>(
      H0, w1, b1, H1, 512, 800, 800, 1008, 1000, H0COLS, 800, 800, 1);
  // H2 = relu(H1 @ w2 + b2): M=512 N=800 K=800
  wmma_gemm_kernel<<<dim3((800 + 127) / 128, 8), 256, 0, stream>>>(
      H1, w2, b2, H2, 512, 800, 800, 800, 800, 800, 800, 800, 1);
  // E = H2 @ we + be: M=512 N=400 K=800
  wmma_gemm_kernel<<<dim3((400 + 127) / 128, 8), 256, 0, stream>>>(
      H2, we, be, E, 512, 400, 400, 800, 800, 800, 400, 400, 0);

  // Stage 3: cosine similarity, 256 outputs
  cosine_kernel<<<32, 256, 0, stream>>>(E, (float*)d_out);
}